// AutoregressiveWaveNO_61744449847315
// MI455X (gfx1250) — compile-verified
//
#include <hip/hip_runtime.h>
#include <math.h>

// ---------------------------------------------------------------------------
// AutoregressiveWaveNO for MI455X (gfx1250, wave32).
// Heavy math (truncated DFT, irfft, pointwise conv) runs on V_WMMA_F32_16X16X4_F32
// keeping full fp32 precision; all step intermediates are L2-resident so the
// sequential 63-step rollout is latency/compute bound -> maximize wave count
// feeding the matrix pipes (split-K DFT) and prefetch the strided operand
// stream (global_prefetch_b8).
// ---------------------------------------------------------------------------

typedef __attribute__((ext_vector_type(2))) float v2f;
typedef __attribute__((ext_vector_type(8))) float v8f;

#define Bc     64
#define NTc    64
#define NXc    2048
#define Dc     16
#define Kc     16
#define HIDc   64
#define NHc    4
#define HDc    16
#define FHc    32
#define MODESc 16
#define NLAc   2
#define NLFc   2
#define NFSc   8
#define NFPc   8
#define FEATc  38           // 2*(1+2*NFS)+4
#define NPc    2458         // 2048 + round(0.2*2048)
#define NSTR   2464         // row stride, multiple of 16, zero-padded tail
#define DFT_CHUNKS 8
#define DFT_COLS   (NSTR / DFT_CHUNKS)   // 308, multiple of 4
#define PIf    3.14159265358979323846f

__device__ __forceinline__ float geluf(float x) {
    return 0.5f * x * (1.0f + erff(x * 0.70710678118654752440f));
}
__device__ __forceinline__ float clamp01(float x) {
    return fminf(fmaxf(x, 0.0f), 1.0f);
}

// fp32 WMMA: D(16x16) += A(16x4) * B(4x16).
// A lane L: M = L%16, holds K = (L/16)*2 + {0,1}.
// B lane L: N = L%16, holds K = (L/16)*2 + {0,1}.
// D vgpr j: lanes 0-15 -> M=j, lanes 16-31 -> M=j+8, N=L%16.
__device__ __forceinline__ v8f wmma_f32(v2f a, v2f b, v8f c) {
    return __builtin_amdgcn_wmma_f32_16x16x4_f32(false, a, false, b, (short)0, c, false, false);
}

// ---------------------------------------------------------------------------
// 1. Segment encoder: feat(B,K,38) -> MLP -> LN -> seg_emb(B,K,64)
// ---------------------------------------------------------------------------
__global__ __launch_bounds__(64)
void seg_encoder_kernel(const float* __restrict__ xs, const float* __restrict__ ks,
                        const float* __restrict__ w1, const float* __restrict__ b1,
                        const float* __restrict__ w2, const float* __restrict__ b2,
                        const float* __restrict__ lng, const float* __restrict__ lnb,
                        float* __restrict__ seg_emb) {
    int b = blockIdx.x;
    int t = threadIdx.x;
    __shared__ float feat[Kc][FEATc];
    __shared__ float h1[Kc][HIDc];
    __shared__ float h2[Kc][HIDc];
    __shared__ float mv[Kc][2];
    if (t < Kc) {
        int k = t;
        float xl = xs[b * (Kc + 1) + k];
        float xr = xs[b * (Kc + 1) + k + 1];
        float w = xr - xl;
        float kv = ks[b * Kc + k];
        float cm = 0.0f;
        for (int j = 0; j <= k; j++)
            cm += ks[b * Kc + j] * (xs[b * (Kc + 1) + j + 1] - xs[b * (Kc + 1) + j]);
        feat[k][0] = xl;
        float f = PIf;
        for (int j = 0; j < NFSc; j++) { feat[k][1 + j] = sinf(xl * f); feat[k][1 + NFSc + j] = cosf(xl * f); f *= 2.0f; }
        feat[k][17] = xr;
        f = PIf;
        for (int j = 0; j < NFSc; j++) { feat[k][18 + j] = sinf(xr * f); feat[k][18 + NFSc + j] = cosf(xr * f); f *= 2.0f; }
        feat[k][34] = w; feat[k][35] = kv; feat[k][36] = kv * (1.0f - kv); feat[k][37] = cm;
    }
    __syncthreads();
    for (int k = 0; k < Kc; k++) {
        float acc = b1[t];
        for (int f2 = 0; f2 < FEATc; f2++) acc += feat[k][f2] * w1[f2 * HIDc + t];
        h1[k][t] = geluf(acc);
    }
    __syncthreads();
    for (int k = 0; k < Kc; k++) {
        float acc = b2[t];
        for (int j = 0; j < HIDc; j++) acc += h1[k][j] * w2[j * HIDc + t];
        h2[k][t] = acc;
    }
    __syncthreads();
    if (t < Kc) {
        float m = 0.0f;
        for (int j = 0; j < HIDc; j++) m += h2[t][j];
        m /= (float)HIDc;
        float v = 0.0f;
        for (int j = 0; j < HIDc; j++) { float d = h2[t][j] - m; v += d * d; }
        v /= (float)HIDc;
        mv[t][0] = m; mv[t][1] = rsqrtf(v + 1e-5f);
    }
    __syncthreads();
    for (int k = 0; k < Kc; k++)
        seg_emb[(b * Kc + k) * HIDc + t] = (h2[k][t] - mv[k][0]) * mv[k][1] * lng[t] + lnb[t];
}

// ---------------------------------------------------------------------------
// 2. Both attention+FFN layers fused; one workgroup per batch, all in LDS.
// ---------------------------------------------------------------------------
__global__ __launch_bounds__(256)
void attn_kernel(const float* __restrict__ wq, const float* __restrict__ wk,
                 const float* __restrict__ wv, const float* __restrict__ wo,
                 const float* __restrict__ bq, const float* __restrict__ bk,
                 const float* __restrict__ bv, const float* __restrict__ bo,
                 const float* __restrict__ ln1g, const float* __restrict__ ln1b,
                 const float* __restrict__ ln2g, const float* __restrict__ ln2b,
                 const float* __restrict__ fw1, const float* __restrict__ fb1,
                 const float* __restrict__ fw2, const float* __restrict__ fb2,
                 const float* __restrict__ pieces_mask, float* __restrict__ seg_emb) {
    int b = blockIdx.x, tid = threadIdx.x;
    __shared__ float E[Kc][HIDc];
    __shared__ float Q[Kc][HIDc], Km[Kc][HIDc], V[Kc][HIDc];
    __shared__ float S[NHc][Kc][Kc];
    __shared__ float O[Kc][HIDc];
    __shared__ float T[Kc][HIDc];
    __shared__ float F1[Kc][4 * HIDc];
    __shared__ float maskk[Kc];
    __shared__ float mv[Kc][2];
    for (int i = tid; i < Kc * HIDc; i += 256) E[i / HIDc][i % HIDc] = seg_emb[(b * Kc) * HIDc + i];
    if (tid < Kc) maskk[tid] = (pieces_mask[b * Kc + tid] <= 0.0f) ? 1.0f : 0.0f;
    __syncthreads();
    if (tid == 0) {
        bool all = true;
        for (int k = 0; k < Kc; k++) if (maskk[k] == 0.0f) { all = false; break; }
        if (all) for (int k = 0; k < Kc; k++) maskk[k] = 0.0f;
    }
    __syncthreads();
    for (int l = 0; l < NLAc; l++) {
        const float* Wq = wq + l * HIDc * HIDc;
        const float* Wk = wk + l * HIDc * HIDc;
        const float* Wv = wv + l * HIDc * HIDc;
        const float* Wo = wo + l * HIDc * HIDc;
        for (int i = tid; i < Kc * HIDc; i += 256) {
            int k = i / HIDc, j = i % HIDc;
            float aq = bq[l * HIDc + j], ak = bk[l * HIDc + j], av = bv[l * HIDc + j];
            for (int c = 0; c < HIDc; c++) {
                float e = E[k][c];
                aq += e * Wq[c * HIDc + j];
                ak += e * Wk[c * HIDc + j];
                av += e * Wv[c * HIDc + j];
            }
            Q[k][j] = aq; Km[k][j] = ak; V[k][j] = av;
        }
        __syncthreads();
        for (int i = tid; i < NHc * Kc * Kc; i += 256) {
            int h = i / (Kc * Kc), q = (i / Kc) % Kc, kk = i % Kc;
            float s = 0.0f;
            for (int d2 = 0; d2 < HDc; d2++) s += Q[q][h * HDc + d2] * Km[kk][h * HDc + d2];
            s *= 0.25f;                        // 1/sqrt(hd), hd=16
            if (maskk[kk] != 0.0f) s = -1e9f;
            S[h][q][kk] = s;
        }
        __syncthreads();
        if (tid < NHc * Kc) {
            int h = tid / Kc, q = tid % Kc;
            float mx = -1e30f;
            for (int kk = 0; kk < Kc; kk++) mx = fmaxf(mx, S[h][q][kk]);
            float sum = 0.0f;
            for (int kk = 0; kk < Kc; kk++) { float e = expf(S[h][q][kk] - mx); S[h][q][kk] = e; sum += e; }
            float inv = 1.0f / sum;
            for (int kk = 0; kk < Kc; kk++) S[h][q][kk] *= inv;
        }
        __syncthreads();
        for (int i = tid; i < Kc * HIDc; i += 256) {
            int k = i / HIDc, j = i % HIDc;
            int h = j / HDc, d2 = j % HDc;
            float acc = 0.0f;
            for (int kk = 0; kk < Kc; kk++) acc += S[h][k][kk] * V[kk][h * HDc + d2];
            O[k][j] = acc;
        }
        __syncthreads();
        for (int i = tid; i < Kc * HIDc; i += 256) {
            int k = i / HIDc, j = i % HIDc;
            float acc = E[k][j] + bo[l * HIDc + j];
            for (int c = 0; c < HIDc; c++) acc += O[k][c] * Wo[c * HIDc + j];
            T[k][j] = acc;
        }
        __syncthreads();
        if (tid < Kc) {
            float m = 0.0f;
            for (int j = 0; j < HIDc; j++) m += T[tid][j];
            m /= (float)HIDc;
            float v = 0.0f;
            for (int j = 0; j < HIDc; j++) { float d = T[tid][j] - m; v += d * d; }
            mv[tid][0] = m; mv[tid][1] = rsqrtf(v / (float)HIDc + 1e-5f);
        }
        __syncthreads();
        for (int i = tid; i < Kc * HIDc; i += 256) {
            int k = i / HIDc, j = i % HIDc;
            E[k][j] = (T[k][j] - mv[k][0]) * mv[k][1] * ln1g[l * HIDc + j] + ln1b[l * HIDc + j];
        }
        __syncthreads();
        for (int i = tid; i < Kc * 4 * HIDc; i += 256) {
            int k = i / (4 * HIDc), m = i % (4 * HIDc);
            float acc = fb1[l * 4 * HIDc + m];
            for (int c = 0; c < HIDc; c++) acc += E[k][c] * fw1[(l * HIDc + c) * (4 * HIDc) + m];
            F1[k][m] = geluf(acc);
        }
        __syncthreads();
        for (int i = tid; i < Kc * HIDc; i += 256) {
            int k = i / HIDc, j = i % HIDc;
            float acc = E[k][j] + fb2[l * HIDc + j];
            for (int m = 0; m < 4 * HIDc; m++) acc += F1[k][m] * fw2[(l * 4 * HIDc + m) * HIDc + j];
            T[k][j] = acc;
        }
        __syncthreads();
        if (tid < Kc) {
            float m = 0.0f;
            for (int j = 0; j < HIDc; j++) m += T[tid][j];
            m /= (float)HIDc;
            float v = 0.0f;
            for (int j = 0; j < HIDc; j++) { float d = T[tid][j] - m; v += d * d; }
            mv[tid][0] = m; mv[tid][1] = rsqrtf(v / (float)HIDc + 1e-5f);
        }
        __syncthreads();
        for (int i = tid; i < Kc * HIDc; i += 256) {
            int k = i / HIDc, j = i % HIDc;
            E[k][j] = (T[k][j] - mv[k][0]) * mv[k][1] * ln2g[l * HIDc + j] + ln2b[l * HIDc + j];
        }
        __syncthreads();
    }
    for (int i = tid; i < Kc * HIDc; i += 256) {
        int k = i / HIDc;
        seg_emb[(b * Kc) * HIDc + i] = E[k][i % HIDc] * pieces_mask[b * Kc + k];
    }
}

// ---------------------------------------------------------------------------
// 3. Boundary-point MLP: concat(seg[li], seg[ri]) -> 64 -> 64 -> LN -> *mask
// ---------------------------------------------------------------------------
__global__ __launch_bounds__(64)
void bp_kernel(const float* __restrict__ seg_emb,
               const float* __restrict__ w1, const float* __restrict__ b1,
               const float* __restrict__ w2, const float* __restrict__ b2,
               const float* __restrict__ lng, const float* __restrict__ lnb,
               const float* __restrict__ disc_mask, float* __restrict__ bp_emb) {
    int b = blockIdx.x, t = threadIdx.x;
    __shared__ float IN[Dc][2 * HIDc];
    __shared__ float H1[Dc][HIDc];
    __shared__ float H2[Dc][HIDc];
    __shared__ float mv[Dc][2];
    for (int i = t; i < Dc * 2 * HIDc; i += 64) {
        int d = i / (2 * HIDc), c = i % (2 * HIDc);
        int kk = (c < HIDc) ? d : ((d + 1 < Kc) ? d + 1 : Kc - 1);
        int cc = (c < HIDc) ? c : c - HIDc;
        IN[d][c] = seg_emb[(b * Kc + kk) * HIDc + cc];
    }
    __syncthreads();
    for (int d = 0; d < Dc; d++) {
        float acc = b1[t];
        for (int c = 0; c < 2 * HIDc; c++) acc += IN[d][c] * w1[c * HIDc + t];
        H1[d][t] = geluf(acc);
    }
    __syncthreads();
    for (int d = 0; d < Dc; d++) {
        float acc = b2[t];
        for (int c = 0; c < HIDc; c++) acc += H1[d][c] * w2[c * HIDc + t];
        H2[d][t] = acc;
    }
    __syncthreads();
    if (t < Dc) {
        float m = 0.0f;
        for (int j = 0; j < HIDc; j++) m += H2[t][j];
        m /= (float)HIDc;
        float v = 0.0f;
        for (int j = 0; j < HIDc; j++) { float d2 = H2[t][j] - m; v += d2 * d2; }
        mv[t][0] = m; mv[t][1] = rsqrtf(v / (float)HIDc + 1e-5f);
    }
    __syncthreads();
    for (int d = 0; d < Dc; d++) {
        float val = (H2[d][t] - mv[d][0]) * mv[d][1] * lng[t] + lnb[t];
        bp_emb[(b * Dc + d) * HIDc + t] = val * disc_mask[b * Dc + d];
    }
}

// ---------------------------------------------------------------------------
// 4. DFT basis tables. basisF[x][0..15]=cos, [16..31]=-sin (rfft e^{-i}).
//    basisI[2m][x]=cos, [2m+1][x]=-sin (for irfft-as-GEMM). Angles reduced
//    via (m*x) mod N for accuracy. Zero for x >= NP (pads WMMA K loop).
// ---------------------------------------------------------------------------
__global__ void basis_kernel(float* __restrict__ basisF, float* __restrict__ basisI) {
    int idx = blockIdx.x * blockDim.x + threadIdx.x;
    if (idx >= NSTR * MODESc) return;
    int x = idx / MODESc, m = idx % MODESc;
    float c = 0.0f, s = 0.0f;
    if (x < NPc) {
        int r = (m * x) % NPc;
        float th = 2.0f * PIf * (float)r / (float)NPc;
        sincosf(th, &s, &c);
    }
    basisF[x * 2 * MODESc + m] = c;
    basisF[x * 2 * MODESc + MODESc + m] = -s;
    basisI[(2 * m) * NSTR + x] = c;
    basisI[(2 * m + 1) * NSTR + x] = -s;
}

// ---------------------------------------------------------------------------
// 5. t=0 outputs + carry init
// ---------------------------------------------------------------------------
__global__ void init_kernel(const float* __restrict__ grid_input, const float* __restrict__ disc,
                            float* __restrict__ state, float* __restrict__ positions,
                            float* __restrict__ out) {
    long idx = (long)blockIdx.x * blockDim.x + threadIdx.x;
    if (idx < (long)Bc * NXc) {
        long b = idx / NXc, x = idx % NXc;
        float v = grid_input[(b * NTc + 0) * NXc + x];
        state[idx] = v;
        out[(b * NTc + 0) * NXc + x] = v;
    }
    if (idx < Bc * Dc) {
        float p = disc[idx * 3 + 0];
        positions[idx] = p;
        out[(long)Bc * NTc * NXc + idx * NTc + 0] = p;
    }
}

// zero the pad-tail columns of an h buffer once (combine never writes x>=NP)
__global__ void zero_tail_kernel(float* __restrict__ h) {
    int idx = blockIdx.x * blockDim.x + threadIdx.x;  // (Bc*FHc) rows x (NSTR-NPc) cols
    int ncol = NSTR - NPc;
    if (idx >= Bc * FHc * ncol) return;
    int row = idx / ncol, col = idx % ncol;
    h[(long)row * NSTR + NPc + col] = 0.0f;
}

// ---------------------------------------------------------------------------
// 6. Per-step: lift + lp/rp channels -> h(B,32,NSTR); zero pad x>=NX
// ---------------------------------------------------------------------------
__global__ __launch_bounds__(256)
void build_h0_kernel(const float* __restrict__ state, const float* __restrict__ positions,
                     const float* __restrict__ disc_mask, const float* __restrict__ x_coords,
                     const float* __restrict__ dt,
                     const float* __restrict__ lift_w, const float* __restrict__ lift_b,
                     float* __restrict__ h) {
    int b = blockIdx.y;
    int x = blockIdx.x * blockDim.x + threadIdx.x;
    if (x >= NSTR) return;
    if (x >= NXc) {
        for (int c = 0; c < FHc; c++) h[(long)(b * FHc + c) * NSTR + x] = 0.0f;
        return;
    }
    float st = state[b * NXc + x];
    float dtv = dt[b];
    float xg = x_coords[(long)b * NTc * NXc + x];
    float lp = 0.0f, rp = 1.0f;
    for (int d = 0; d < Dc; d++) {
        float p = positions[b * Dc + d];
        bool valid = disc_mask[b * Dc + d] > 0.0f;
        if (valid && p <= xg) lp = fmaxf(lp, p);
        if (valid && p > xg) rp = fminf(rp, p);
    }
    float f2 = xg - lp, f3 = rp - xg;
    for (int c = 0; c < FHc; c++) {
        float v = lift_b[c] + lift_w[c * 4 + 0] * st + lift_w[c * 4 + 1] * dtv
                + lift_w[c * 4 + 2] * f2 + lift_w[c * 4 + 3] * f3;
        h[(long)(b * FHc + c) * NSTR + x] = v;
    }
}

// ---------------------------------------------------------------------------
// 7a. Split-K truncated forward DFT via WMMA.
//     Grid (B, 4 tiles, 8 chunks) = 2048 waves; each wave runs 77 f32-WMMA
//     K-steps over its 308-column slice and writes a partial 16x16 tile.
//     global_prefetch_b8 pulls the strided h stream + basis stream ahead.
// ---------------------------------------------------------------------------
__global__ __launch_bounds__(32)
void dft_part_kernel(const float* __restrict__ h, const float* __restrict__ basisF,
                     float* __restrict__ part) {
    int b = blockIdx.x;
    int tile = blockIdx.y;              // 0..3
    int chunk = blockIdx.z;             // 0..7
    int ct = tile & 1, mt = tile >> 1;  // channel tile / {cos,-sin} tile
    int L = threadIdx.x;
    int lan = L & 15;
    int koff = (L >> 4) * 2;
    int x0 = chunk * DFT_COLS;
    const float* hrow = h + (long)(b * FHc + ct * 16 + lan) * NSTR + x0 + koff;
    const float* brow = basisF + (long)(x0 + koff) * (2 * MODESc) + mt * MODESc + lan;
    v8f acc = {0.f, 0.f, 0.f, 0.f, 0.f, 0.f, 0.f, 0.f};
    for (int kb = 0; kb < DFT_COLS; kb += 4) {
        if ((kb & 63) == 0) {
            __builtin_prefetch(hrow + kb + 128, 0, 1);
            __builtin_prefetch(brow + (long)(kb + 64) * 2 * MODESc, 0, 1);
        }
        v2f a, bt;
        a.x = hrow[kb];
        a.y = hrow[kb + 1];
        bt.x = brow[(long)kb * 2 * MODESc];
        bt.y = brow[(long)kb * 2 * MODESc + 2 * MODESc];
        acc = wmma_f32(a, bt, acc);
    }
    long pbase = (long)((b * 4 + tile) * DFT_CHUNKS + chunk) * 256;
    for (int j = 0; j < 8; j++) {
        int M = j + (L >> 4) * 8;
        part[pbase + M * 16 + lan] = acc[j];
    }
}

// 7b. Reduce the 8 partial tiles into hf[b,c,m,{re,im}]
__global__ void dft_reduce_kernel(const float* __restrict__ part, float* __restrict__ hf) {
    int idx = blockIdx.x * blockDim.x + threadIdx.x;
    if (idx >= Bc * FHc * MODESc * 2) return;
    int t2 = idx;
    int reim = t2 & 1; t2 >>= 1;
    int m = t2 % MODESc; t2 /= MODESc;
    int c = t2 % FHc; t2 /= FHc;
    int b = t2;
    int tile = reim * 2 + (c >> 4);     // mt*2 + ct
    int M = c & 15;
    float sum = 0.0f;
    for (int ch = 0; ch < DFT_CHUNKS; ch++)
        sum += part[(long)((b * 4 + tile) * DFT_CHUNKS + ch) * 256 + M * 16 + m];
    hf[((long)(b * FHc + c) * MODESc + m) * 2 + reim] = sum;
}

// ---------------------------------------------------------------------------
// 8. Spectral complex matmul (tiny): sf[b,o,m] = sum_i (wr+j wi)[i,o,m]*hf[b,i,m]
// ---------------------------------------------------------------------------
__global__ void specmm_kernel(const float* __restrict__ hf, const float* __restrict__ wr,
                              const float* __restrict__ wi, float* __restrict__ sf, int l) {
    int idx = blockIdx.x * blockDim.x + threadIdx.x;
    if (idx >= Bc * FHc * MODESc) return;
    int m = idx % MODESc;
    int o = (idx / MODESc) % FHc;
    int b = idx / (MODESc * FHc);
    float ar = 0.0f, ai = 0.0f;
    for (int i = 0; i < FHc; i++) {
        float hr = hf[((long)(b * FHc + i) * MODESc + m) * 2 + 0];
        float hi = hf[((long)(b * FHc + i) * MODESc + m) * 2 + 1];
        float r = wr[((l * FHc + i) * FHc + o) * MODESc + m];
        float w = wi[((l * FHc + i) * FHc + o) * MODESc + m];
        ar += hr * r - hi * w;
        ai += hr * w + hi * r;
    }
    sf[((long)(b * FHc + o) * MODESc + m) * 2 + 0] = ar;
    sf[((long)(b * FHc + o) * MODESc + m) * 2 + 1] = ai;
}

// A-matrix element for the irfft-as-GEMM extension (k interleaves re/im rows)
__device__ __forceinline__ float sf_a(const float* __restrict__ sf, int b, int o, int k, float invN) {
    int m = k >> 1;
    float scale = (m == 0) ? ((k & 1) ? 0.0f : invN) : (2.0f * invN);
    return sf[((long)(b * FHc + o) * MODESc + m) * 2 + (k & 1)] * scale;
}

// ---------------------------------------------------------------------------
// 9. Combine: hout = irfft(sf) + h @ pw_w + pw_b, optional gelu.
//    One wave per (b, 16-col x-tile) computing all 32 output channels:
//    K=32 pointwise + K=32 irfft -> 32 f32 WMMAs per tile.
// ---------------------------------------------------------------------------
__global__ __launch_bounds__(32)
void combine_kernel(const float* __restrict__ h, const float* __restrict__ sf,
                    const float* __restrict__ basisI,
                    const float* __restrict__ pw_w, const float* __restrict__ pw_b,
                    float* __restrict__ hout, int l, int apply_gelu) {
    int b = blockIdx.x;
    int x0 = blockIdx.y * 16;
    int L = threadIdx.x;
    int lan = L & 15;
    int hi2 = L >> 4;
    v8f acc0 = {0.f, 0.f, 0.f, 0.f, 0.f, 0.f, 0.f, 0.f};
    v8f acc1 = {0.f, 0.f, 0.f, 0.f, 0.f, 0.f, 0.f, 0.f};
    const float* W = pw_w + l * FHc * FHc;
    // pointwise conv: out[o,x] += sum_i h[i,x] * W[i,o]
    for (int kb = 0; kb < FHc; kb += 4) {
        int k0 = kb + hi2 * 2;
        v2f a0, a1, bt;
        a0.x = W[k0 * FHc + lan];        a0.y = W[(k0 + 1) * FHc + lan];
        a1.x = W[k0 * FHc + 16 + lan];   a1.y = W[(k0 + 1) * FHc + 16 + lan];
        bt.x = h[(long)(b * FHc + k0) * NSTR + x0 + lan];
        bt.y = h[(long)(b * FHc + k0 + 1) * NSTR + x0 + lan];
        acc0 = wmma_f32(a0, bt, acc0);
        acc1 = wmma_f32(a1, bt, acc1);
    }
    // irfft: out[o,x] += sum_k Aext[o,k] * basisI[k,x]
    const float invN = 1.0f / (float)NPc;
    for (int kb = 0; kb < 2 * MODESc; kb += 4) {
        int k0 = kb + hi2 * 2;
        v2f a0, a1, bt;
        a0.x = sf_a(sf, b, lan, k0, invN);       a0.y = sf_a(sf, b, lan, k0 + 1, invN);
        a1.x = sf_a(sf, b, 16 + lan, k0, invN);  a1.y = sf_a(sf, b, 16 + lan, k0 + 1, invN);
        bt.x = basisI[(long)k0 * NSTR + x0 + lan];
        bt.y = basisI[(long)(k0 + 1) * NSTR + x0 + lan];
        acc0 = wmma_f32(a0, bt, acc0);
        acc1 = wmma_f32(a1, bt, acc1);
    }
    int xcol = x0 + lan;
    for (int j = 0; j < 8; j++) {
        int o0 = j + hi2 * 8;
        int o1 = 16 + o0;
        float v0 = acc0[j] + pw_b[l * FHc + o0];
        float v1 = acc1[j] + pw_b[l * FHc + o1];
        if (apply_gelu) { v0 = geluf(v0); v1 = geluf(v1); }
        if (xcol < NPc) {
            hout[(long)(b * FHc + o0) * NSTR + xcol] = v0;
            hout[(long)(b * FHc + o1) * NSTR + xcol] = v1;
        }
    }
}

// ---------------------------------------------------------------------------
// 10. Projection + state update + write output slab t
// ---------------------------------------------------------------------------
__global__ __launch_bounds__(256)
void proj_state_kernel(const float* __restrict__ h, const float* __restrict__ proj_w,
                       const float* __restrict__ proj_b,
                       float* __restrict__ state, float* __restrict__ out, int t) {
    long idx = (long)blockIdx.x * blockDim.x + threadIdx.x;
    if (idx >= (long)Bc * NXc) return;
    long b = idx / NXc, x = idx % NXc;
    float acc = proj_b[0];
    for (int c = 0; c < FHc; c++) acc += h[(long)(b * FHc + c) * NSTR + x] * proj_w[c];
    float s = clamp01(state[idx] + tanhf(acc));
    state[idx] = s;
    out[(b * NTc + t) * NXc + x] = s;
}

// ---------------------------------------------------------------------------
// 11. Trajectory MLP: positions update + write position slab t
// ---------------------------------------------------------------------------
__global__ __launch_bounds__(256)
void traj_kernel(const float* __restrict__ bp_emb, const float* __restrict__ disc_mask,
                 const float* __restrict__ dt,
                 const float* __restrict__ w1, const float* __restrict__ b1,
                 const float* __restrict__ w2, const float* __restrict__ b2,
                 float* __restrict__ positions, float* __restrict__ out, int t) {
    int idx = blockIdx.x * blockDim.x + threadIdx.x;
    if (idx >= Bc * Dc) return;
    int b = idx / Dc;
    float pos = positions[idx];
    float in[82];
    for (int c = 0; c < HIDc; c++) in[c] = bp_emb[(long)idx * HIDc + c];
    in[64] = pos;
    float f = PIf;
    for (int j = 0; j < NFPc; j++) { in[65 + j] = sinf(pos * f); in[65 + NFPc + j] = cosf(pos * f); f *= 2.0f; }
    in[81] = dt[b];
    float outv = b2[0];
    for (int j = 0; j < HIDc; j++) {
        float acc = b1[j];
        #pragma unroll
        for (int c = 0; c < 82; c++) acc += in[c] * w1[c * HIDc + j];
        outv += geluf(acc) * w2[j];
    }
    float delta = tanhf(outv) * 0.1f;
    pos = clamp01(pos + delta) * disc_mask[idx];
    positions[idx] = pos;
    out[(long)Bc * NTc * NXc + (long)idx * NTc + t] = pos;
}

// ---------------------------------------------------------------------------
// Host orchestration
// ---------------------------------------------------------------------------
extern "C" void kernel_launch(void* const* d_in, const int* in_sizes, int n_in,
                              void* d_out, int out_size, void* d_ws, size_t ws_size,
                              hipStream_t stream) {
    (void)in_sizes; (void)n_in; (void)out_size; (void)ws_size;
    const float* grid_input  = (const float*)d_in[0];
    const float* xs          = (const float*)d_in[1];
    const float* ks          = (const float*)d_in[2];
    const float* pieces_mask = (const float*)d_in[3];
    const float* disc        = (const float*)d_in[4];
    const float* disc_mask   = (const float*)d_in[5];
    const float* x_coords    = (const float*)d_in[6];
    const float* dt          = (const float*)d_in[7];
    const float* se_w1 = (const float*)d_in[8],  *se_b1 = (const float*)d_in[9];
    const float* se_w2 = (const float*)d_in[10], *se_b2 = (const float*)d_in[11];
    const float* se_ln_g = (const float*)d_in[12], *se_ln_b = (const float*)d_in[13];
    const float* attn_wq = (const float*)d_in[14], *attn_wk = (const float*)d_in[15];
    const float* attn_wv = (const float*)d_in[16], *attn_wo = (const float*)d_in[17];
    const float* attn_bq = (const float*)d_in[18], *attn_bk = (const float*)d_in[19];
    const float* attn_bv = (const float*)d_in[20], *attn_bo = (const float*)d_in[21];
    const float* ln1g = (const float*)d_in[22], *ln1b = (const float*)d_in[23];
    const float* ln2g = (const float*)d_in[24], *ln2b = (const float*)d_in[25];
    const float* ffn_w1 = (const float*)d_in[26], *ffn_b1 = (const float*)d_in[27];
    const float* ffn_w2 = (const float*)d_in[28], *ffn_b2 = (const float*)d_in[29];
    const float* bp_w1 = (const float*)d_in[30], *bp_b1 = (const float*)d_in[31];
    const float* bp_w2 = (const float*)d_in[32], *bp_b2 = (const float*)d_in[33];
    const float* bp_ln_g = (const float*)d_in[34], *bp_ln_b = (const float*)d_in[35];
    const float* lift_w = (const float*)d_in[36], *lift_b = (const float*)d_in[37];
    const float* spec_wr = (const float*)d_in[38], *spec_wi = (const float*)d_in[39];
    const float* pw_w = (const float*)d_in[40], *pw_b = (const float*)d_in[41];
    const float* proj_w = (const float*)d_in[42], *proj_b = (const float*)d_in[43];
    const float* traj_w1 = (const float*)d_in[44], *traj_b1 = (const float*)d_in[45];
    const float* traj_w2 = (const float*)d_in[46], *traj_b2 = (const float*)d_in[47];
    float* out = (float*)d_out;

    // workspace layout
    float* ws = (float*)d_ws;
    size_t off = 0;
    float* seg_emb = ws + off;   off += (size_t)Bc * Kc * HIDc;
    float* bp_emb  = ws + off;   off += (size_t)Bc * Dc * HIDc;
    float* state   = ws + off;   off += (size_t)Bc * NXc;
    float* positions = ws + off; off += (size_t)Bc * Dc;
    float* hf = ws + off;        off += (size_t)Bc * FHc * MODESc * 2;
    float* sf = ws + off;        off += (size_t)Bc * FHc * MODESc * 2;
    float* basisF = ws + off;    off += (size_t)NSTR * 2 * MODESc;
    float* basisI = ws + off;    off += (size_t)2 * MODESc * NSTR;
    float* dftp = ws + off;      off += (size_t)Bc * 4 * DFT_CHUNKS * 256;
    float* hA = ws + off;        off += (size_t)Bc * FHc * NSTR;
    float* hB = ws + off;        off += (size_t)Bc * FHc * NSTR;

    // one-time setup
    seg_encoder_kernel<<<Bc, 64, 0, stream>>>(xs, ks, se_w1, se_b1, se_w2, se_b2,
                                              se_ln_g, se_ln_b, seg_emb);
    attn_kernel<<<Bc, 256, 0, stream>>>(attn_wq, attn_wk, attn_wv, attn_wo,
                                        attn_bq, attn_bk, attn_bv, attn_bo,
                                        ln1g, ln1b, ln2g, ln2b,
                                        ffn_w1, ffn_b1, ffn_w2, ffn_b2,
                                        pieces_mask, seg_emb);
    bp_kernel<<<Bc, 64, 0, stream>>>(seg_emb, bp_w1, bp_b1, bp_w2, bp_b2,
                                     bp_ln_g, bp_ln_b, disc_mask, bp_emb);
    basis_kernel<<<(NSTR * MODESc + 255) / 256, 256, 0, stream>>>(basisF, basisI);
    init_kernel<<<((Bc * NXc) + 255) / 256, 256, 0, stream>>>(grid_input, disc, state, positions, out);
    // hB pad-tail columns [NP, NSTR) must be zero (combine never writes them)
    zero_tail_kernel<<<(Bc * FHc * (NSTR - NPc) + 255) / 256, 256, 0, stream>>>(hB);

    for (int t = 1; t < NTc; t++) {
        build_h0_kernel<<<dim3((NSTR + 255) / 256, Bc), 256, 0, stream>>>(
            state, positions, disc_mask, x_coords, dt, lift_w, lift_b, hA);
        // FNO layer 0: hA -> hB (with gelu)
        dft_part_kernel<<<dim3(Bc, 4, DFT_CHUNKS), 32, 0, stream>>>(hA, basisF, dftp);
        dft_reduce_kernel<<<(Bc * FHc * MODESc * 2 + 255) / 256, 256, 0, stream>>>(dftp, hf);
        specmm_kernel<<<(Bc * FHc * MODESc + 255) / 256, 256, 0, stream>>>(hf, spec_wr, spec_wi, sf, 0);
        combine_kernel<<<dim3(Bc, NSTR / 16), 32, 0, stream>>>(hA, sf, basisI, pw_w, pw_b, hB, 0, 1);
        // FNO layer 1: hB -> hA (no gelu)
        dft_part_kernel<<<dim3(Bc, 4, DFT_CHUNKS), 32, 0, stream>>>(hB, basisF, dftp);
        dft_reduce_kernel<<<(Bc * FHc * MODESc * 2 + 255) / 256, 256, 0, stream>>>(dftp, hf);
        specmm_kernel<<<(Bc * FHc * MODESc + 255) / 256, 256, 0, stream>>>(hf, spec_wr, spec_wi, sf, 1);
        combine_kernel<<<dim3(Bc, NSTR / 16), 32, 0, stream>>>(hB, sf, basisI, pw_w, pw_b, hA, 1, 0);
        proj_state_kernel<<<((Bc * NXc) + 255) / 256, 256, 0, stream>>>(hA, proj_w, proj_b, state, out, t);
        traj_kernel<<<(Bc * Dc + 255) / 256, 256, 0, stream>>>(bp_emb, disc_mask, dt,
                                                               traj_w1, traj_b1, traj_w2, traj_b2,
                                                               positions, out, t);
    }
}